// SplineNet_24550033063953
// MI455X (gfx1250) — compile-verified
//
#include <hip/hip_runtime.h>
#include <hip/hip_bf16.h>

#define AS1 __attribute__((address_space(1)))
#define AS3 __attribute__((address_space(3)))

typedef int   v4i __attribute__((vector_size(16)));
typedef float v4f __attribute__((ext_vector_type(4)));

#define BETA_LN2 0.6931471805599453f   // softplus beta = ln(2)

// ---------------------------------------------------------------------------
// d_ws layout (bytes):
//   [   0,   64) : kx[16]   (float)  knot x-coordinates (bin edges)
//   [  64,  544) : rec[15][8] (float) per-bin record:
//                  {x0, 1/dx, y0, dy*s, dy*d0, s, d0+d1-2s, pad}
//   [ 544, 1056) : lut[512] (u8)    bin index of each cell's LEFT edge
// Total: 1056 bytes, staged once per block into LDS by the main kernel
// through the CDNA5 async global->LDS path (ASYNCcnt).
// ---------------------------------------------------------------------------

__device__ __forceinline__ float rqs_eval(float xc,
                                          const float* __restrict__ s_kx,
                                          const float* __restrict__ s_rec,
                                          const unsigned char* __restrict__ s_lut) {
  // quantized-x LUT start, then exact fixup (rarely taken: ~3% of lanes)
  int u = (int)(xc * 512.0f);            // x*512 is exact (power-of-2 scale)
  u = (u < 0) ? 0 : ((u > 511) ? 511 : u);
  int b = (int)s_lut[u];
  while (b < 14 && xc >= s_kx[b + 1]) ++b;

  const v4f* rp = (const v4f*)(s_rec + (b << 3));  // 32B records, 16B aligned
  v4f r0 = rp[0];   // {x0, 1/dx, y0, dy*s}
  v4f r1 = rp[1];   // {dy*d0, s, c, pad}

  float xi  = (xc - r0.x) * r0.y;
  float t   = xi * (1.0f - xi);
  float num = __builtin_fmaf(r0.w, xi * xi, r1.x * t);   // dy*(s*xi^2 + d0*xi*(1-xi))
  float den = __builtin_fmaf(r1.z, t, r1.y);             // s + (d0+d1-2s)*xi*(1-xi)
  return __builtin_fmaf(num, __builtin_amdgcn_rcpf(den), r0.z);  // y0 + num/den
}

__global__ void __launch_bounds__(256)
rqs_forward_kernel(const float* __restrict__ x, float* __restrict__ out,
                   const float* __restrict__ ws, int n) {
  __shared__ __align__(16) unsigned char smem[1088];
  const float*         s_kx  = (const float*)smem;
  const float*         s_rec = (const float*)(smem + 64);
  const unsigned char* s_lut = smem + 544;

  // Stage the 1056-byte table into LDS via the CDNA5 async global->LDS path
  // (ASYNCcnt), falling back to plain loads if the builtin is unavailable.
#if __has_builtin(__builtin_amdgcn_global_load_async_to_lds_b128)
  if (threadIdx.x < 66) {   // 66 * 16B = 1056B
    __builtin_amdgcn_global_load_async_to_lds_b128(
        (AS1 v4i*)((const unsigned char*)ws + threadIdx.x * 16),
        (AS3 v4i*)(smem + threadIdx.x * 16),
        0, 0);
  }
#if __has_builtin(__builtin_amdgcn_s_wait_asynccnt)
  __builtin_amdgcn_s_wait_asynccnt(0);
#else
  asm volatile("s_wait_asynccnt 0" ::: "memory");
#endif
#else
  {
    const unsigned int* g = (const unsigned int*)ws;
    unsigned int*       l = (unsigned int*)smem;
    for (int i = threadIdx.x; i < 264; i += blockDim.x) l[i] = g[i];
  }
#endif
  __syncthreads();

  int tid      = blockIdx.x * blockDim.x + threadIdx.x;
  int nthreads = gridDim.x * blockDim.x;
  int n4       = n >> 2;

  const v4f* x4 = (const v4f*)x;
  v4f*       o4 = (v4f*)out;

  for (int i = tid; i < n4; i += nthreads) {
    // Streaming data (268MB total > 192MB L2): non-temporal policy on both
    // sides so the single-pass traffic does not thrash L2.
    v4f v = __builtin_nontemporal_load(x4 + i);   // global_load_b128 (NT)
    v4f r;
    r.x = rqs_eval(v.x, s_kx, s_rec, s_lut);
    r.y = rqs_eval(v.y, s_kx, s_rec, s_lut);
    r.z = rqs_eval(v.z, s_kx, s_rec, s_lut);
    r.w = rqs_eval(v.w, s_kx, s_rec, s_lut);
    __builtin_nontemporal_store(r, o4 + i);       // global_store_b128 (NT)
  }

  // scalar tail (n is a multiple of 4 for this problem, but stay general)
  int t = (n4 << 2) + tid;
  if (t < n) out[t] = rqs_eval(x[t], s_kx, s_rec, s_lut);
}

// ---------------------------------------------------------------------------

__global__ void __launch_bounds__(128)
build_table_kernel(const float* __restrict__ wx,
                   const float* __restrict__ wy,
                   const float* __restrict__ wd,
                   float* __restrict__ ws) {
  __shared__ float kx[16];
  __shared__ float ky[16];
  __shared__ float kd[16];

  if (threadIdx.x == 0) {
    // knots_x = [0, cumsum(softmax(wx))]
    float m = -1e30f;
    for (int k = 0; k < 15; ++k) m = fmaxf(m, wx[k]);
    float e[15], sum = 0.0f;
    for (int k = 0; k < 15; ++k) { e[k] = expf(wx[k] - m); sum += e[k]; }
    float inv = 1.0f / sum, acc = 0.0f;
    kx[0] = 0.0f;
    for (int k = 0; k < 15; ++k) { acc += e[k] * inv; kx[k + 1] = acc; }

    // knots_y = [0, cumsum(softmax(wy))]
    m = -1e30f;
    for (int k = 0; k < 15; ++k) m = fmaxf(m, wy[k]);
    sum = 0.0f;
    for (int k = 0; k < 15; ++k) { e[k] = expf(wy[k] - m); sum += e[k]; }
    inv = 1.0f / sum; acc = 0.0f;
    ky[0] = 0.0f;
    for (int k = 0; k < 15; ++k) { acc += e[k] * inv; ky[k + 1] = acc; }

    // knots_d = softplus(beta*wd)/beta  (stable logaddexp form)
    for (int k = 0; k < 16; ++k) {
      float bx = BETA_LN2 * wd[k];
      kd[k] = (fmaxf(bx, 0.0f) + log1pf(expf(-fabsf(bx)))) * (1.0f / BETA_LN2);
    }
  }
  __syncthreads();

  if (threadIdx.x < 16) ws[threadIdx.x] = kx[threadIdx.x];

  if (threadIdx.x < 15) {
    int b = threadIdx.x;
    float x0 = kx[b], dx = kx[b + 1] - x0;
    float y0 = ky[b], dy = ky[b + 1] - y0;
    float s  = dy / dx;
    float d0 = kd[b], d1 = kd[b + 1];
    float* r = ws + 16 + b * 8;
    r[0] = x0;
    r[1] = 1.0f / dx;
    r[2] = y0;
    r[3] = dy * s;                 // dy*s      -> num  = dys*xi^2 + dyd0*t
    r[4] = dy * d0;                // dy*d0
    r[5] = s;                      // den = c*t + s
    r[6] = d0 + d1 - 2.0f * s;     // c
    r[7] = 0.0f;
  }

  // LUT: bin index at each cell's left edge.  i/512 is exact in fp32, and the
  // runtime uses the identical comparisons, so lut[u] <= bin(x) always holds
  // and the fixup loop in the main kernel restores exact searchsorted("right").
  unsigned char* lut = (unsigned char*)ws + 544;
  for (int i = threadIdx.x; i < 512; i += blockDim.x) {
    float xv = (float)i * (1.0f / 512.0f);
    int b = 0;
    for (int k = 1; k <= 14; ++k) b += (kx[k] <= xv) ? 1 : 0;
    lut[i] = (unsigned char)b;
  }
}

// ---------------------------------------------------------------------------

extern "C" void kernel_launch(void* const* d_in, const int* in_sizes, int n_in,
                              void* d_out, int out_size, void* d_ws, size_t ws_size,
                              hipStream_t stream) {
  const float* x  = (const float*)d_in[0];
  const float* wx = (const float*)d_in[1];
  const float* wy = (const float*)d_in[2];
  const float* wd = (const float*)d_in[3];
  float* out = (float*)d_out;
  float* ws  = (float*)d_ws;
  int n = in_sizes[0];

  build_table_kernel<<<1, 128, 0, stream>>>(wx, wy, wd, ws);

  int n4 = n >> 2;
  int threads = 256;
  long long want = (long long)(n4 > 0 ? n4 : 1);
  int blocks = (int)((want + threads - 1) / threads);
  if (blocks < 1) blocks = 1;
  rqs_forward_kernel<<<blocks, threads, 0, stream>>>(x, out, ws, n);
}